// MultiHeadAttentionBlock_40999757808217
// MI455X (gfx1250) — compile-verified
//
#include <hip/hip_runtime.h>
#include <stdint.h>

// MI455X / gfx1250, wave32. Reference reduces (see round-0 analysis) to:
//   V = x @ wv + bv                          (only needed where kv_idx[b]==0)
//   A[b,:] = bf16_rne( kv_idx[b]==0 ? V[b,:] : kv_value[b, pos0(b), :] )
//   Y = A @ wo + bo
// Both GEMMs: M=16 (batch), K=1024, N=1024, weights row-major K x N.
// Bandwidth-bound (~8.4 MB moved, 67 MFLOP) -> exact-f32 WMMA 16x16x4,
// async memory->LDS staging for the A matrix.

typedef float v2f __attribute__((ext_vector_type(2)));
typedef float v8f __attribute__((ext_vector_type(8)));
typedef int   v4i __attribute__((ext_vector_type(4)));

#define FDIM   1024
#define BATCH  16
#define CCTX   4096
#define NWAVES 8
#define KCHUNK (FDIM / NWAVES)   // 128 K-values per wave
#define APAD   4                 // row stride 1028 floats: 16B-aligned rows AND
                                 // conflict-free 32-lane column reads (stride%64==4)

#if defined(__has_builtin)
#  if __has_builtin(__builtin_amdgcn_global_load_async_to_lds_b128)
#    define HAVE_ASYNC_LDS 1
#  endif
#  if __has_builtin(__builtin_amdgcn_s_wait_asynccnt)
#    define HAVE_WAIT_ASYNC 1
#  endif
#endif

typedef __attribute__((address_space(1))) v4i* gv4i_ptr;  // global (HIP "__device__")
typedef __attribute__((address_space(3))) v4i* lv4i_ptr;  // LDS

struct LdsLayout {
  float A[BATCH][FDIM + APAD];
  float part[NWAVES][256];      // per-wave 16x16 f32 partials, WMMA lane order
};

__device__ __forceinline__ void wait_async0() {
#if defined(HAVE_WAIT_ASYNC)
  __builtin_amdgcn_s_wait_asynccnt(0);
#else
  asm volatile("s_wait_asynccnt 0x0" ::: "memory");
#endif
}

// 16B async copy global -> LDS (falls back to a plain load/store pair).
__device__ __forceinline__ void copy16_async(const float* src, float* lds_dst) {
#if defined(HAVE_ASYNC_LDS)
  __builtin_amdgcn_global_load_async_to_lds_b128(
      (gv4i_ptr)(v4i*)(void*)const_cast<float*>(src),
      (lv4i_ptr)(v4i*)(void*)lds_dst,
      /*offset=*/0, /*cpol=*/0);
#else
  *(float4*)lds_dst = *(const float4*)src;
#endif
}

__device__ __forceinline__ float bf16_rne(float x) {
  // round-to-nearest-even to bf16, back to f32 (inputs are finite randn)
  uint32_t u = __float_as_uint(x);
  u = (u + 0x7FFFu + ((u >> 16) & 1u)) & 0xFFFF0000u;
  return __uint_as_float(u);
}

// One 16-column tile of C[16,1024] = A(16,1024, staged raw in LDS) @ B + bias.
// 8 waves split K; each wave issues V_WMMA_F32_16X16X4_F32; LDS tree-reduce.
// ROUND_A applies the reference's bf16 cast to A elements at fragment build.
template <bool ROUND_A>
__device__ __forceinline__ void gemm_tile(LdsLayout* lds,
                                          const float* __restrict__ Bmat,
                                          const float* __restrict__ bias,
                                          float* __restrict__ out, int n0) {
  const int tid  = threadIdx.x;
  const int wave = tid >> 5;
  const int lane = tid & 31;
  const int nloc = lane & 15;            // n within tile (and M row for A frag)
  const int koff = (lane >> 4) << 1;     // lanes 0-15 -> K+0..1, lanes 16-31 -> K+2..3
  const int k0   = wave * KCHUNK;

  v8f acc = {0.f, 0.f, 0.f, 0.f, 0.f, 0.f, 0.f, 0.f};
  const float* Bp = Bmat + (size_t)k0 * FDIM + n0 + nloc;

#pragma unroll 4
  for (int kk = 0; kk < KCHUNK; kk += 4) {
    const int k = k0 + kk + koff;
    v2f a, b;
    // A-fragment: 32-bit 16x4 layout (ISA 7.12.2): lane m holds A[m, koff..koff+1]
    a.x = ROUND_A ? bf16_rne(lds->A[nloc][k]) : lds->A[nloc][k];
    a.y = ROUND_A ? bf16_rne(lds->A[nloc][k + 1]) : lds->A[nloc][k + 1];
    // B-fragment: rows k (v0) and k+1 (v1), 16 contiguous columns per half-wave
    const float* bp = Bp + (size_t)(kk + koff) * FDIM;
    b.x = bp[0];
    b.y = bp[FDIM];
    acc = __builtin_amdgcn_wmma_f32_16x16x4_f32(
        /*neg_a=*/false, a, /*neg_b=*/false, b,
        /*c_mod=*/(short)0, acc, /*reuse_a=*/false, /*reuse_b=*/false);
  }

#pragma unroll
  for (int r = 0; r < 8; ++r) lds->part[wave][r * 32 + lane] = acc[r];
  __syncthreads();

  // 256 threads: one (m,n) each. C layout: VGPR r, lanes0-15 -> M=r, lanes16-31 -> M=r+8.
  const int m   = tid >> 4;
  const int n   = tid & 15;
  const int idx = ((m & 7) << 5) + ((m >> 3) << 4) + n;
  float s = bias[n0 + n];
#pragma unroll
  for (int w = 0; w < NWAVES; ++w) s += lds->part[w][idx];
  out[(size_t)m * FDIM + n0 + n] = s;
}

// Kernel 1: V = x @ wv + bv, but only if some batch actually needs it (kv_idx==0).
__global__ void __launch_bounds__(256)
vproj_kernel(const float* __restrict__ x, const int* __restrict__ kv_idx,
             const float* __restrict__ wv, const float* __restrict__ bv,
             float* __restrict__ Vout) {
  bool need = false;
  for (int b = 0; b < BATCH; ++b) need |= (kv_idx[b] == 0);
  if (!need) return;   // uniform branch: skip 4 MB of wv traffic when dead

  __shared__ LdsLayout lds;
  // Stage x[16,1024] into LDS via async 16B copies (4096 chunks, 16/thread).
  for (int c = threadIdx.x; c < (BATCH * FDIM) / 4; c += 256) {
    const int b = c >> 8, k4 = (c & 255) << 2;
    copy16_async(x + ((size_t)c << 2), &lds.A[b][k4]);
  }
  wait_async0();
  __syncthreads();

  gemm_tile<false>(&lds, wv, bv, Vout, blockIdx.x * 16);
}

// Kernel 2: gather the surviving value row per batch (raw f32, async to LDS),
// bf16-round at fragment time (matches reference attn bf16), Y = A @ wo + bo.
__global__ void __launch_bounds__(256)
oproj_kernel(const int* __restrict__ kv_idx, const float* __restrict__ kv_value,
             const float* __restrict__ V, const float* __restrict__ wo,
             const float* __restrict__ bo, float* __restrict__ y) {
  __shared__ LdsLayout lds;
  for (int c = threadIdx.x; c < (BATCH * FDIM) / 4; c += 256) {
    const int b = c >> 8, k4 = (c & 255) << 2;
    const int idx = kv_idx[b];
    // rotated-cache position 0: start of ring = (idx+1)%C once the ring wrapped
    const int p = (idx + 1 > CCTX) ? ((idx + 1) & (CCTX - 1)) : 0;
    const float* row = (idx == 0)
                           ? V + ((size_t)b << 10)
                           : kv_value + (((size_t)b * CCTX + p) << 10);
    copy16_async(row + k4, &lds.A[b][k4]);
  }
  wait_async0();
  __syncthreads();

  gemm_tile<true>(&lds, wo, bo, y, blockIdx.x * 16);
}

extern "C" void kernel_launch(void* const* d_in, const int* in_sizes, int n_in,
                              void* d_out, int out_size, void* d_ws, size_t ws_size,
                              hipStream_t stream) {
  // setup_inputs order: x, mask, kv_idx, kv_key, kv_value, wq, bq, wk, bk, wv, bv, wo, bo
  const float* x        = (const float*)d_in[0];
  const int*   kv_idx   = (const int*)d_in[2];
  const float* kv_value = (const float*)d_in[4];
  const float* wv       = (const float*)d_in[9];
  const float* bv       = (const float*)d_in[10];
  const float* wo       = (const float*)d_in[11];
  const float* bo       = (const float*)d_in[12];
  (void)in_sizes; (void)n_in; (void)out_size; (void)ws_size;

  float* V = (float*)d_ws;      // 16x1024 f32 scratch (64 KB)
  float* y = (float*)d_out;     // 16x1x1024 f32

  vproj_kernel<<<64, 256, 0, stream>>>(x, kv_idx, wv, bv, V);
  oproj_kernel<<<64, 256, 0, stream>>>(kv_idx, kv_value, V, wo, bo, y);
}